// AttentionScore_73899207295521
// MI455X (gfx1250) — compile-verified
//
#include <hip/hip_runtime.h>
#include <math.h>

typedef __attribute__((ext_vector_type(2))) float v2f;
typedef __attribute__((ext_vector_type(8))) float v8f;

#define B_DIM 2048
#define G_DIM 1024
#define H_DIM 256
#define C_CLIP 10.0f
#define SCALE 0.0625f   /* 1/sqrt(256) */

// One block per batch b. 256 threads = 8 wave32s; wave w covers g in
// [w*128, w*128+128) as 8 groups of 16 g's. Per group, accumulate
// u[g] = sum_h q[h]*k[g,h] with V_WMMA_F32_16X16X4_F32:
//   A (16x4)  = query chunk broadcast to all 16 rows
//   B (4x16)  = 16 key rows, K-chunk of 4
//   D (16x16) = u[g0+n] replicated over rows m
// A-layout (ISA 7.12.2, 32-bit A 16x4): lane<16 holds K=0,1; lane>=16 K=2,3.
// B mirrors: lane<16 holds K=0,1 for N=lane; lane>=16 K=2,3 for N=lane-16.
__global__ __launch_bounds__(256) void attn_score_wmma(
    const float* __restrict__ q,              // [B,1,H]
    const float* __restrict__ k,              // [B,G,H]
    const unsigned char* __restrict__ mask,   // [B,G] (bool, 1 byte)
    float* __restrict__ out)                  // [B,1,G]
{
    __shared__ float lds_q[H_DIM];

    const int b    = blockIdx.x;
    const int tid  = threadIdx.x;
    const int wave = tid >> 5;
    const int lane = tid & 31;

    // stage this batch's query row (1 KB) into LDS
    lds_q[tid] = q[(size_t)b * H_DIM + tid];
    __syncthreads();

    const int n    = lane & 15;        // B-matrix column (g within group)
    const int ksel = (lane >> 4) * 2;  // K pair owned by this half-wave: {0,2}

    const int g0 = wave * 128;
    // per-lane key pointer: row (g0+n), starting at this lane's K pair
    const float* kp = k + ((size_t)b * G_DIM + g0 + n) * H_DIM + ksel;

    const v8f vzero = {0.f, 0.f, 0.f, 0.f, 0.f, 0.f, 0.f, 0.f};
    v8f acc[8];
    #pragma unroll
    for (int j = 0; j < 8; ++j) acc[j] = vzero;

    #pragma unroll 1
    for (int h0 = 0; h0 < H_DIM; h0 += 4) {
        // A fragment: q[h0+ksel], q[h0+ksel+1] (identical for every row m)
        v2f a = *(const v2f*)&lds_q[h0 + ksel];
        #pragma unroll
        for (int j = 0; j < 8; ++j) {
            // B fragment: key[g0+16j+n, h0+ksel .. +1]  (8B, 8B-aligned)
            v2f bv = *(const v2f*)(kp + j * (16 * H_DIM) + h0);
            acc[j] = __builtin_amdgcn_wmma_f32_16x16x4_f32(
                /*neg_a=*/false, a, /*neg_b=*/false, bv,
                /*c_mod=*/(short)0, acc[j],
                /*reuse_a=*/false, /*reuse_b=*/false);
        }
    }

    // Extraction: lane<16 holds D[0,lane] in c[0] == u[g0+16j+lane].
    // (Rows of D are identical since A's rows are identical.)
    if (lane < 16) {
        #pragma unroll
        for (int j = 0; j < 8; ++j) {
            const int    g   = g0 + j * 16 + lane;
            const size_t idx = (size_t)b * G_DIM + g;
            float u     = acc[j][0] * SCALE;
            float logit = tanhf(u) * C_CLIP;
            if (mask[idx]) logit = -INFINITY;
            out[idx] = logit;   // coalesced 64B store per group
        }
    }
}

extern "C" void kernel_launch(void* const* d_in, const int* in_sizes, int n_in,
                              void* d_out, int out_size, void* d_ws, size_t ws_size,
                              hipStream_t stream) {
    const float*         q    = (const float*)d_in[0];
    const float*         kk   = (const float*)d_in[1];
    const unsigned char* mask = (const unsigned char*)d_in[2];
    float*               out  = (float*)d_out;
    (void)in_sizes; (void)n_in; (void)d_ws; (void)ws_size; (void)out_size;

    dim3 grid(B_DIM);
    dim3 block(256);
    hipLaunchKernelGGL(attn_score_wmma, grid, block, 0, stream, q, kk, mask, out);
}